// RoiAlign_42649025249502
// MI455X (gfx1250) — compile-verified
//
#include <hip/hip_runtime.h>
#include <stdint.h>
#include <stddef.h>

// RoIAlign / crop_and_resize, bilinear, 7x7 pool.
// features: [B, 64, 64, 256] f32   rois: [B*N, 4] f32 (y1,x1,y2,x2 normalized)
// out:      [B*N, 7, 7, 256] f32
//
// Roofline: ~200 MB of streaming output writes -> ~9 us at 23.3 TB/s HBM.
// Features (8.4 MB) stay resident in the 192 MB L2, so the 4x gather read
// amplification is L2 traffic, not HBM. One block (8 wave32) per ROI;
// wave-uniform sampling math scalarized via readfirstlane so gathers use the
// saddr+voffset form; 128-bit gathers; non-temporal 128-bit stores; async DMA
// broadcast of the ROI descriptor into LDS (gfx1250 ASYNCcnt path).

typedef float v4f __attribute__((ext_vector_type(4)));

#define POOL_ 7
#define H_    64
#define W_    64
#define C_    256
#define CG_   (C_ / 4)   // 64 float4 channel groups

__device__ __forceinline__ int rfl_i(int x) {
    return __builtin_amdgcn_readfirstlane(x);
}
__device__ __forceinline__ float rfl_f(float x) {
    return __int_as_float(__builtin_amdgcn_readfirstlane(__float_as_int(x)));
}

__global__ __launch_bounds__(256) void roi_align_nhwc_kernel(
    const float* __restrict__ feat,   // [B,H,W,C]
    const float* __restrict__ rois,   // [B*N,4]
    float*       __restrict__ out,    // [B*N,49,C]
    int nPerBatch)
{
    __shared__ float s_roi[4];

    const int roi = blockIdx.x;
    const int tid = threadIdx.x;

    // --- Stage the 16B ROI descriptor into LDS with one async DMA copy ------
    // (gfx1250 async global->LDS path; ASYNCcnt-tracked, lane 0 only)
    if (tid == 0) {
        uint32_t lds_off = (uint32_t)(uintptr_t)(&s_roi[0]);
        const float* g = rois + (size_t)roi * 4;
        asm volatile("global_load_async_to_lds_b128 %0, %1, off"
                     :: "v"(lds_off), "v"(g)
                     : "memory");
    }
    asm volatile("s_wait_asynccnt 0" ::: "memory");
    __syncthreads();

    const float ry1 = s_roi[0];
    const float rx1 = s_roi[1];
    const float ry2 = s_roi[2];
    const float rx2 = s_roi[3];

    const int b = roi / nPerBatch;
    const float* __restrict__ fbase = feat + (size_t)b * (H_ * W_ * C_);

    const int cg   = tid & (CG_ - 1);   // channel group 0..63
    const int sl   = tid >> 6;          // sample-lane 0..3 (uniform per wave)
    const int coff = cg * 4;

    // Per-ROI uniform sampling parameters
    const float stepy = (ry2 - ry1) * (1.0f / (POOL_ - 1)) * (float)(H_ - 1);
    const float stepx = (rx2 - rx1) * (1.0f / (POOL_ - 1)) * (float)(W_ - 1);
    const float basey = ry1 * (float)(H_ - 1);
    const float basex = rx1 * (float)(W_ - 1);

    float* __restrict__ obase = out + (size_t)roi * (POOL_ * POOL_ * C_) + coff;

    // Thread group sl covers samples s = sl, sl+4, ..., <49 (13 iters max).
    int py = 0;
    int px = sl;

#pragma unroll
    for (int k = 0; k < 13; ++k) {
        const int s = sl + 4 * k;
        if (s < POOL_ * POOL_) {
            // --- y axis (wave-uniform values) ---
            const float ysf = fmaf((float)py, stepy, basey);
            float y0f = floorf(ysf);
            y0f = fminf(fmaxf(y0f, 0.0f), (float)(H_ - 1));
            const int   y0  = (int)y0f;
            const int   y1i = min(y0 + 1, H_ - 1);

            // --- x axis (wave-uniform values) ---
            const float xsf = fmaf((float)px, stepx, basex);
            float x0f = floorf(xsf);
            x0f = fminf(fmaxf(x0f, 0.0f), (float)(W_ - 1));
            const int   x0  = (int)x0f;
            const int   x1i = min(x0 + 1, W_ - 1);

            // Scalarize: pixel offsets -> SGPRs (loads become saddr+voffset),
            // weights -> SGPR operands of the fma chain.
            const int o00 = rfl_i(y0  * W_ + x0 ) * C_;
            const int o01 = rfl_i(y0  * W_ + x1i) * C_;
            const int o10 = rfl_i(y1i * W_ + x0 ) * C_;
            const int o11 = rfl_i(y1i * W_ + x1i) * C_;
            const float wy = rfl_f(ysf - y0f);
            const float wx = rfl_f(xsf - x0f);

            // --- 4 corner gathers: 128-bit loads, hot in L2 ---
            const v4f f00 = *(const v4f*)(fbase + o00 + coff);
            const v4f f01 = *(const v4f*)(fbase + o01 + coff);
            const v4f f10 = *(const v4f*)(fbase + o10 + coff);
            const v4f f11 = *(const v4f*)(fbase + o11 + coff);

            // --- bilinear (same association order as the reference) ---
            const v4f top = f00 + (f01 - f00) * wx;
            const v4f bot = f10 + (f11 - f10) * wx;
            const v4f res = top + (bot - top) * wy;

            // --- streaming write: non-temporal 128-bit store (TH=NT) ---
            __builtin_nontemporal_store(res, (v4f*)(obase + (size_t)s * C_));

            // incremental sample-index update (avoids s/7 division)
            px += 4;
            if (px >= POOL_) { px -= POOL_; ++py; }
        }
    }
}

extern "C" void kernel_launch(void* const* d_in, const int* in_sizes, int n_in,
                              void* d_out, int out_size, void* d_ws, size_t ws_size,
                              hipStream_t stream) {
    (void)n_in; (void)out_size; (void)d_ws; (void)ws_size;

    const float* feat = (const float*)d_in[0];
    const float* rois = (const float*)d_in[1];
    float*       out  = (float*)d_out;

    const int totalRois = in_sizes[1] / 4;             // B*N
    int B = in_sizes[0] / (H_ * W_ * C_);
    if (B < 1) B = 1;
    const int nPerBatch = totalRois / B;

    roi_align_nhwc_kernel<<<totalRois, 256, 0, stream>>>(feat, rois, out, nPerBatch);
}